// LSMPool_72988674228491
// MI455X (gfx1250) — compile-verified
//
#include <hip/hip_runtime.h>
#include <hip/hip_bf16.h>

#define SIZE  4096
#define STEPS 512
#define WROW  8192   // W is 4096 x 8192 row-major
#define THRESH 10.0f
#define NBLK  16     // persistent-kernel blocks (4096 threads total, 1/neuron)

typedef __attribute__((ext_vector_type(2))) float v2f;
typedef __attribute__((ext_vector_type(8))) float v8f;

// ---------------------------------------------------------------------------
// Coalesced 32x32 LDS-tiled transpose of one 4096x4096 half of W (f32).
// outT[s*4096 + o] = W[o*8192 + colOfs + s]
// ---------------------------------------------------------------------------
__global__ void __launch_bounds__(256)
snn_transpose_f32(const float* __restrict__ W, float* __restrict__ outT, int colOfs) {
    __shared__ float tile[32][33];
    const int bx = blockIdx.x * 32;
    const int by = blockIdx.y * 32;
    const int tx = threadIdx.x & 31;
    const int ty = threadIdx.x >> 5;
#pragma unroll
    for (int i = 0; i < 32; i += 8)
        tile[ty + i][tx] = W[(size_t)(by + ty + i) * WROW + colOfs + bx + tx];
    __syncthreads();
#pragma unroll
    for (int i = 0; i < 32; i += 8)
        outT[(size_t)(bx + ty + i) * SIZE + by + tx] = tile[tx][ty + i];
}

// ---------------------------------------------------------------------------
// cur_x = x(512x4096) @ Wx^T via V_WMMA_F32_16X16X4_F32 (full f32: the LIF
// threshold dynamics are chaotic, so bf16 weights would diverge from the
// reference over 512 steps; the GEMM is only 17 GFLOP anyway).
// ---------------------------------------------------------------------------
__global__ void __launch_bounds__(256)
snn_input_gemm_wmma(const float* __restrict__ A,   // x: 512 x 4096 row-major
                    const float* __restrict__ BT,  // WxT: 4096(k) x 4096(n)
                    float* __restrict__ C) {       // cur_x staged into mem_rec
    const int K = SIZE, N = SIZE;
    const int lane  = threadIdx.x & 31;
    const int wave  = threadIdx.x >> 5;
    const int mtile = blockIdx.y << 4;
    const int ntile = (blockIdx.x * 8 + wave) << 4;
    const int mn = lane & 15;
    const int kh = (lane >> 4) << 1;

    const float* ap = A  + (size_t)(mtile + mn) * K + kh;
    const float* bp = BT + (size_t)kh * N + ntile + mn;

    v8f c = {};
#pragma unroll 8
    for (int kk = 0; kk < K; kk += 4) {
        v2f a, b;
        a.x = ap[0];
        a.y = ap[1];
        b.x = bp[0];
        b.y = bp[N];
        c = __builtin_amdgcn_wmma_f32_16x16x4_f32(
                false, a, false, b, (short)0, c, false, false);
        ap += 4;
        bp += (size_t)4 * N;
    }

    const int hi = lane >> 4;
#pragma unroll
    for (int r = 0; r < 8; ++r)
        C[(size_t)(mtile + r + 8 * hi) * N + ntile + mn] = c[r];
}

// ---------------------------------------------------------------------------
__global__ void snn_init_counter(int* c) {
    if (threadIdx.x == 0 && blockIdx.x == 0) *c = 0;
}

// ---------------------------------------------------------------------------
// Persistent LIF recurrence: one launch instead of 512. 16 blocks x 256
// threads (one thread per neuron); membrane potential lives in a register.
// Per step: async-copy the previous spike row (16KB) to LDS
// (global_load_async_to_lds_b128 / s_wait_asynccnt), ordered compaction of
// active indices (ascending j -> deterministic f32 order), accumulate only
// active WsT rows (L2-resident), threshold+reset, then a device-scope
// release/acquire spin barrier before the next step.
// ---------------------------------------------------------------------------
__global__ void __launch_bounds__(256)
snn_lif_persistent(const float* __restrict__ WsT,   // 4096(j) x 4096(o)
                   float* __restrict__ spk_rec,     // 512 x 4096
                   float* __restrict__ mem_rec,     // 512 x 4096 (holds cur_x)
                   int* barrier_cnt) {
    __shared__ __align__(16) float sspk[SIZE];      // 16 KB
    __shared__ unsigned short slist[SIZE];          //  8 KB
    __shared__ int sscan[256];
    __shared__ int stotal;

    const int tid = threadIdx.x;
    const int o   = blockIdx.x * 256 + tid;
    const float* col = WsT + o;

    float mem = 0.0f;

    for (int t = 0; t < STEPS; ++t) {
        float acc = mem + mem_rec[(size_t)t * SIZE + o];   // + cur_x[t,o]

        if (t > 0) {
            // ---- async copy of spk_rec[t-1][0:4096] into LDS (64B/thread) ----
            {
                const char* g = (const char*)(spk_rec + (size_t)(t - 1) * SIZE)
                                + tid * 64;
                unsigned int l = (unsigned int)(size_t)(void*)sspk
                                 + (unsigned int)(tid * 64);
#pragma unroll
                for (int i = 0; i < 4; ++i) {
                    unsigned long long ga = (unsigned long long)(size_t)(g + i * 16);
                    unsigned int       la = l + i * 16;
                    asm volatile("global_load_async_to_lds_b128 %0, %1, off"
                                 :: "v"(la), "v"(ga) : "memory");
                }
                asm volatile("s_wait_asynccnt 0x0" ::: "memory");
            }
            __syncthreads();

            // ---- ordered compaction: 16 j's per thread + prefix scan ----
            const int jb = tid * 16;
            int c = 0;
#pragma unroll
            for (int i = 0; i < 16; ++i) c += (sspk[jb + i] != 0.0f) ? 1 : 0;
            sscan[tid] = c;
            __syncthreads();
            for (int ofs = 1; ofs < 256; ofs <<= 1) {
                int v = (tid >= ofs) ? sscan[tid - ofs] : 0;
                __syncthreads();
                sscan[tid] += v;
                __syncthreads();
            }
            int pos = sscan[tid] - c;                  // exclusive prefix
            if (tid == 255) stotal = sscan[255];
#pragma unroll
            for (int i = 0; i < 16; ++i)
                if (sspk[jb + i] != 0.0f)
                    slist[pos++] = (unsigned short)(jb + i);
            __syncthreads();

            // ---- accumulate active rows, ascending j (deterministic) ----
            const int nact = stotal;
            for (int k = 0; k < nact; ++k) {
                const int j = slist[k];
                if (k + 8 < nact)
                    __builtin_prefetch(col + (size_t)slist[k + 8] * SIZE, 0, 0);
                acc += col[(size_t)j * SIZE];          // coalesced, L2-hot
            }
        }

        const float spk = (acc - THRESH > 0.0f) ? 1.0f : 0.0f;
        mem = acc - spk * THRESH;
        spk_rec[(size_t)t * SIZE + o] = spk;
        mem_rec[(size_t)t * SIZE + o] = mem;

        // ---- grid-wide barrier: make step-t spikes visible everywhere ----
        __syncthreads();
        if (tid == 0) {
            __hip_atomic_fetch_add(barrier_cnt, 1, __ATOMIC_RELEASE,
                                   __HIP_MEMORY_SCOPE_AGENT);
            const int target = NBLK * (t + 1);
            while (__hip_atomic_load(barrier_cnt, __ATOMIC_ACQUIRE,
                                     __HIP_MEMORY_SCOPE_AGENT) < target)
                __builtin_amdgcn_s_sleep(1);
        }
        __syncthreads();
    }
}

// ---------------------------------------------------------------------------
extern "C" void kernel_launch(void* const* d_in, const int* in_sizes, int n_in,
                              void* d_out, int out_size, void* d_ws, size_t ws_size,
                              hipStream_t stream) {
    (void)in_sizes; (void)n_in; (void)out_size; (void)ws_size;

    const float* x = (const float*)d_in[0];   // 512 x 4096
    const float* W = (const float*)d_in[1];   // 4096 x 8192

    float* spk_rec = (float*)d_out;                       // 512 x 4096
    float* mem_rec = spk_rec + (size_t)STEPS * SIZE;      // 512 x 4096

    float* WxT = (float*)d_ws;                            // 64 MB
    float* WsT = WxT + (size_t)SIZE * SIZE;               // 64 MB
    int*   cnt = (int*)(WsT + (size_t)SIZE * SIZE);

    dim3 tgrid(SIZE / 32, SIZE / 32);
    snn_transpose_f32<<<tgrid, 256, 0, stream>>>(W, WxT, 0);
    snn_transpose_f32<<<tgrid, 256, 0, stream>>>(W, WsT, SIZE);

    dim3 ggrid(SIZE / (16 * 8), STEPS / 16);              // (32, 32)
    snn_input_gemm_wmma<<<ggrid, 256, 0, stream>>>(x, WxT, mem_rec);

    snn_init_counter<<<1, 64, 0, stream>>>(cnt);
    snn_lif_persistent<<<NBLK, 256, 0, stream>>>(WsT, spk_rec, mem_rec, cnt);
}